// SCC_67886253080785
// MI455X (gfx1250) — compile-verified
//
#include <hip/hip_runtime.h>
#include <math.h>

// ---------------------------------------------------------------------------
// Problem constants
// ---------------------------------------------------------------------------
#define BATCH   4
#define IMG     192
#define CH      60
#define NPIX    (BATCH * IMG * IMG)        // 147456
#define NH      6
#define HD      5
#define HS      16
#define WS      16
#define LWIN    256                        // tokens per window
#define NWH     (IMG / HS)                 // 12
#define NWIN    (BATCH * NWH * NWH)        // 576
#define MLEN    64                         // MH*MW

typedef __attribute__((ext_vector_type(2))) float v2f;
typedef __attribute__((ext_vector_type(8))) float v8f;

#if defined(__HIP_DEVICE_COMPILE__) && __has_builtin(__builtin_amdgcn_wmma_f32_16x16x4_f32)
#define HAVE_WMMA_F32 1
#else
#define HAVE_WMMA_F32 0
#endif

// D = A(16x4) * B(4x16) + C  in exact f32 (V_WMMA_F32_16X16X4_F32).
// Fragment layout per CDNA5 ISA 7.12.2:
//   A: lane m=lane&15; a.x=A[m][lane<16?0:2], a.y=A[m][lane<16?1:3]
//   B: lane n=lane&15; b.x=B[lane<16?0:2][n], b.y=B[lane<16?1:3][n]
//   C/D: c[i] = D[i + (lane>=16?8:0)][lane&15]
__device__ __forceinline__ v8f wmma4(v2f a, v2f b, v8f c) {
#if HAVE_WMMA_F32
  return __builtin_amdgcn_wmma_f32_16x16x4_f32(false, a, false, b, (short)0, c,
                                               false, false);
#else
  const int lane = (int)(threadIdx.x & 31u);
  const int n = lane & 15;
  const float bb0 = __shfl(b.x, n);
  const float bb1 = __shfl(b.y, n);
  const float bb2 = __shfl(b.x, n + 16);
  const float bb3 = __shfl(b.y, n + 16);
#pragma unroll
  for (int i = 0; i < 8; i++) {
    const int m = i + ((lane & 16) ? 8 : 0);
    const float a0 = __shfl(a.x, m);
    const float a1 = __shfl(a.y, m);
    const float a2 = __shfl(a.x, m + 16);
    const float a3 = __shfl(a.y, m + 16);
    c[i] += a0 * bb0 + a1 * bb1 + a2 * bb2 + a3 * bb3;
  }
  return c;
#endif
}

// qv is NHWC (B,192,192,60).  window wi, token l, channel c -> flat index
__device__ __forceinline__ int qv_index(int wi, int l, int c) {
  const int b  = wi / (NWH * NWH);
  const int th = (wi / NWH) % NWH;
  const int tw = wi % NWH;
  const int hh = th * HS + (l >> 4);
  const int ww = tw * WS + (l & 15);
  return ((b * IMG + hh) * IMG + ww) * CH + c;
}

// ---------------------------------------------------------------------------
// Small RefDFE kernels (low-flop stages, VALU)
// ---------------------------------------------------------------------------
__global__ void k_c1(const float* __restrict__ x, const float* __restrict__ w,
                     const float* __restrict__ bias, float* __restrict__ o) {
  const int p = blockIdx.x * blockDim.x + threadIdx.x;
  if (p >= NPIX) return;
  const float* xr = x + p * CH;
  float acc[12];
#pragma unroll
  for (int oc = 0; oc < 12; oc++) acc[oc] = bias[oc];
  for (int ic = 0; ic < CH; ic++) {
    const float xv = xr[ic];
#pragma unroll
    for (int oc = 0; oc < 12; oc++) acc[oc] += xv * w[oc * CH + ic];
  }
#pragma unroll
  for (int oc = 0; oc < 12; oc++) {
    const float v = acc[oc];
    o[p * 12 + oc] = v >= 0.f ? v : 0.2f * v;
  }
}

__global__ void k_c2(const float* __restrict__ c1, const float* __restrict__ w,
                     const float* __restrict__ bias, float* __restrict__ o) {
  const int p = blockIdx.x * blockDim.x + threadIdx.x;
  if (p >= NPIX) return;
  const int b = p / (IMG * IMG);
  const int rem = p % (IMG * IMG);
  const int hh = rem / IMG, ww = rem % IMG;
  float acc[12];
#pragma unroll
  for (int oc = 0; oc < 12; oc++) acc[oc] = bias[oc];
  for (int ky = 0; ky < 3; ky++) {
    const int y = hh + ky - 1;
    if ((unsigned)y >= (unsigned)IMG) continue;
    for (int kx = 0; kx < 3; kx++) {
      const int xw = ww + kx - 1;
      if ((unsigned)xw >= (unsigned)IMG) continue;
      const float* row = c1 + ((b * IMG + y) * IMG + xw) * 12;
      for (int ic = 0; ic < 12; ic++) {
        const float v = row[ic];
#pragma unroll
        for (int oc = 0; oc < 12; oc++)
          acc[oc] += v * w[((oc * 12 + ic) * 3 + ky) * 3 + kx];
      }
    }
  }
#pragma unroll
  for (int oc = 0; oc < 12; oc++) {
    const float v = acc[oc];
    o[p * 12 + oc] = v >= 0.f ? v : 0.2f * v;
  }
}

// dfe written into columns 0..59 of the stride-120 "cat" buffer
__global__ void k_dfe(const float* __restrict__ x, const float* __restrict__ c2,
                      const float* __restrict__ w3, const float* __restrict__ b3,
                      const float* __restrict__ wl, const float* __restrict__ bl,
                      float* __restrict__ cat) {
  const int idx = blockIdx.x * blockDim.x + threadIdx.x;
  if (idx >= NPIX * CH) return;
  const int p = idx / CH, oc = idx % CH;
  float a = b3[oc];
  const float* c2r = c2 + p * 12;
  for (int ic = 0; ic < 12; ic++) a += c2r[ic] * w3[oc * 12 + ic];
  float li = bl[oc];
  const float* xr = x + p * CH;
  for (int ic = 0; ic < CH; ic++) li += xr[ic] * wl[oc * CH + ic];
  cat[p * 120 + oc] = a * li;
}

// ---------------------------------------------------------------------------
// 3x3 conv 60->60 as WMMA implicit GEMM.
// M = 16-pixel row tile (tiles never cross an image row: 192 = 12*16),
// N = 60 output channels (4 tiles, weights zero-padded to 64 rows in LDS),
// K = 9 taps x 60 channels, walked tap-major: per tap exactly 15 K=4 steps.
// The y-border test is uniform per wave (whole tap skipped); the x-border
// test is a per-lane value select on a clamped (always valid) address.
// ---------------------------------------------------------------------------
template <bool NCHW_IN, bool LRELU>
__global__ __launch_bounds__(256) void k_conv3x3(const float* __restrict__ in,
                                                 const float* __restrict__ w,
                                                 const float* __restrict__ bias,
                                                 float* __restrict__ out,
                                                 int ostride, int ooff) {
  __shared__ float w_s[64 * 60];  // per-tap weight slice, rows 60..63 zero
  const int tid = threadIdx.x;
  const int lane = tid & 31, wave = tid >> 5;
  const int halfHi = (lane >= 16) ? 1 : 0;
  const int mlane = lane & 15;
  const int tile = blockIdx.x * 8 + wave;
  const int p0 = tile * 16;
  const int b = p0 / (IMG * IMG);
  const int rem = p0 % (IMG * IMG);
  const int row = rem / IMG;
  const int col0 = rem % IMG;
  const v8f vzero = {0.f, 0.f, 0.f, 0.f, 0.f, 0.f, 0.f, 0.f};
  v8f acc[4];
#pragma unroll
  for (int nt = 0; nt < 4; nt++) acc[nt] = vzero;

  for (int tap = 0; tap < 9; tap++) {
    const int ky = tap / 3, kx = tap % 3;
    for (int t = tid; t < 64 * 60; t += 256) {
      const int n = t / 60, ic = t % 60;
      const int nn = (n < 60) ? n : 0;  // clamp address, select value
      w_s[t] = (n < 60) ? w[(nn * 60 + ic) * 9 + tap] : 0.f;
    }
    __syncthreads();
    const int y = row + ky - 1;
    if ((unsigned)y < (unsigned)IMG) {  // uniform per wave
      const int xw = col0 + mlane + kx - 1;
      const bool xv = (unsigned)xw < (unsigned)IMG;
      const int xwc = xv ? xw : 0;
      int base;
      if (NCHW_IN)
        base = (b * CH) * (IMG * IMG) + y * IMG + xwc;  // + ic*IMG*IMG
      else
        base = ((b * IMG + y) * IMG + xwc) * CH;        // + ic
      for (int ks = 0; ks < 15; ks++) {
        const int ic0 = ks * 4 + (halfHi ? 2 : 0);
        float a0, a1;
        if (NCHW_IN) {
          a0 = xv ? in[base + ic0 * (IMG * IMG)] : 0.f;
          a1 = xv ? in[base + (ic0 + 1) * (IMG * IMG)] : 0.f;
        } else {
          a0 = xv ? in[base + ic0] : 0.f;
          a1 = xv ? in[base + ic0 + 1] : 0.f;
        }
        v2f a = {a0, a1};
#pragma unroll
        for (int nt = 0; nt < 4; nt++) {
          const int n = nt * 16 + mlane;
          v2f bb = {w_s[n * 60 + ic0], w_s[n * 60 + ic0 + 1]};
          acc[nt] = wmma4(a, bb, acc[nt]);
        }
      }
    }
    __syncthreads();
  }
#pragma unroll
  for (int nt = 0; nt < 4; nt++) {
    const int n = nt * 16 + mlane;
    if (n < 60) {
      const float bs = bias[n];
#pragma unroll
      for (int i = 0; i < 8; i++) {
        const int m = i + (halfHi ? 8 : 0);
        float v = acc[nt][i] + bs;
        if (LRELU) v = v >= 0.f ? v : 0.2f * v;
        out[(p0 + m) * ostride + ooff + n] = v;
      }
    }
  }
}

// ---------------------------------------------------------------------------
// Gate 1x1 conv (K=120) as WMMA GEMM + sigmoid blend -> qv (NHWC)
// ---------------------------------------------------------------------------
__global__ __launch_bounds__(256) void k_gate(const float* __restrict__ cat,
                                              const float* __restrict__ wg,
                                              const float* __restrict__ bg,
                                              float* __restrict__ qv) {
  __shared__ float w_s[64 * 120];  // rows 60..63 zero
  __shared__ float bg_s[64];
  const int tid = threadIdx.x;
  for (int t = tid; t < 64 * 120; t += 256) {
    const int n = t / 120, k = t % 120;
    const int nn = (n < 60) ? n : 0;
    w_s[t] = (n < 60) ? wg[nn * 120 + k] : 0.f;
  }
  if (tid < 64) bg_s[tid] = (tid < 60) ? bg[tid] : 0.f;
  __syncthreads();
  const int lane = tid & 31, wave = tid >> 5;
  const int halfHi = (lane >= 16) ? 1 : 0;
  const int mlane = lane & 15;
  const int p0 = (blockIdx.x * 8 + wave) * 16;
  const v8f vzero = {0.f, 0.f, 0.f, 0.f, 0.f, 0.f, 0.f, 0.f};
  v8f acc[4];
#pragma unroll
  for (int nt = 0; nt < 4; nt++) acc[nt] = vzero;
  const int arow = (p0 + mlane) * 120;
  for (int ks = 0; ks < 30; ks++) {
    const int k0 = ks * 4 + (halfHi ? 2 : 0);
    v2f a = {cat[arow + k0], cat[arow + k0 + 1]};
#pragma unroll
    for (int nt = 0; nt < 4; nt++) {
      const int n = nt * 16 + mlane;
      v2f bb = {w_s[n * 120 + k0], w_s[n * 120 + k0 + 1]};
      acc[nt] = wmma4(a, bb, acc[nt]);
    }
  }
#pragma unroll
  for (int nt = 0; nt < 4; nt++) {
    const int n = nt * 16 + mlane;
    if (n < 60) {
#pragma unroll
      for (int i = 0; i < 8; i++) {
        const int m = i + (halfHi ? 8 : 0);
        const int p = p0 + m;
        float g = acc[nt][i] + bg_s[n];
        g = 1.0f / (1.0f + expf(-g));
        const float dv = cat[p * 120 + n];
        const float av = cat[p * 120 + 60 + n];
        qv[p * CH + n] = dv * g + av * (1.0f - g);
      }
    }
  }
}

// ---------------------------------------------------------------------------
// Relative-position bias MLP (tiny)
// ---------------------------------------------------------------------------
__device__ __forceinline__ void ln3_relu(const float* __restrict__ g,
                                         const float* __restrict__ be, float h0,
                                         float h1, float h2, float& t0, float& t1,
                                         float& t2) {
  const float m = (h0 + h1 + h2) * (1.f / 3.f);
  const float d0 = h0 - m, d1 = h1 - m, d2 = h2 - m;
  const float v = (d0 * d0 + d1 * d1 + d2 * d2) * (1.f / 3.f);
  const float inv = rsqrtf(v + 1e-5f);
  t0 = d0 * inv * g[0] + be[0]; t0 = t0 > 0.f ? t0 : 0.f;
  t1 = d1 * inv * g[1] + be[1]; t1 = t1 > 0.f ? t1 : 0.f;
  t2 = d2 * inv * g[2] + be[2]; t2 = t2 > 0.f ? t2 : 0.f;
}

__global__ __launch_bounds__(1024) void k_ptab(
    const float* __restrict__ wpp, const float* __restrict__ bpp,
    const float* __restrict__ g1, const float* __restrict__ be1,
    const float* __restrict__ w1, const float* __restrict__ b1,
    const float* __restrict__ g2, const float* __restrict__ be2,
    const float* __restrict__ w2, const float* __restrict__ b2,
    const float* __restrict__ g3, const float* __restrict__ be3,
    const float* __restrict__ w3, const float* __restrict__ b3,
    float* __restrict__ ptab) {
  const int r = threadIdx.x;
  if (r >= 961) return;
  const float dy = (float)(r / 31 - 15);
  const float dx = (float)(r % 31 - 15);
  float h0 = dy * wpp[0] + dx * wpp[1] + bpp[0];
  float h1 = dy * wpp[2] + dx * wpp[3] + bpp[1];
  float h2 = dy * wpp[4] + dx * wpp[5] + bpp[2];
  float t0, t1, t2;
  ln3_relu(g1, be1, h0, h1, h2, t0, t1, t2);
  h0 = t0 * w1[0] + t1 * w1[1] + t2 * w1[2] + b1[0];
  h1 = t0 * w1[3] + t1 * w1[4] + t2 * w1[5] + b1[1];
  h2 = t0 * w1[6] + t1 * w1[7] + t2 * w1[8] + b1[2];
  ln3_relu(g2, be2, h0, h1, h2, t0, t1, t2);
  h0 = t0 * w2[0] + t1 * w2[1] + t2 * w2[2] + b2[0];
  h1 = t0 * w2[3] + t1 * w2[4] + t2 * w2[5] + b2[1];
  h2 = t0 * w2[6] + t1 * w2[7] + t2 * w2[8] + b2[2];
  ln3_relu(g3, be3, h0, h1, h2, t0, t1, t2);
#pragma unroll
  for (int o = 0; o < 6; o++)
    ptab[r * 6 + o] = t0 * w3[o * 3] + t1 * w3[o * 3 + 1] + t2 * w3[o * 3 + 2] + b3[o];
}

__global__ void k_rpb(const float* __restrict__ ptab, float* __restrict__ rpb) {
  const int idx = blockIdx.x * blockDim.x + threadIdx.x;
  if (idx >= NH * LWIN * MLEN) return;
  const int m = idx & 63;
  const int l = (idx >> 6) & 255;
  const int h = idx >> 14;
  const int r1 = l >> 4, c1 = l & 15;
  const int mh = m >> 3, mw = m & 7;
  float s = 0.f;
#pragma unroll
  for (int rh = 0; rh < 2; rh++)
#pragma unroll
    for (int rw = 0; rw < 2; rw++) {
      const int r2 = mh * 2 + rh, c2 = mw * 2 + rw;
      const int rpi = (r1 - r2 + 15) * 31 + (c1 - c2 + 15);
      s += ptab[rpi * 6 + h];
    }
  rpb[idx] = s * 0.25f;
}

// vp[(wi*6+h)*320 + m*5 + d] : spatially pooled V
__global__ void k_vp(const float* __restrict__ qv, const float* __restrict__ wsl,
                     const float* __restrict__ bsl, float* __restrict__ vp) {
  const int idx = blockIdx.x * blockDim.x + threadIdx.x;
  if (idx >= NWIN * NH * MLEN * HD) return;
  const int d = idx % 5;
  const int m = (idx / 5) % 64;
  const int h = (idx / 320) % 6;
  const int wi = idx / 1920;
  const int mh = m >> 3, mw = m & 7;
  float s = bsl[0];
#pragma unroll
  for (int rh = 0; rh < 2; rh++)
#pragma unroll
    for (int rw = 0; rw < 2; rw++) {
      const int l = (mh * 2 + rh) * 16 + (mw * 2 + rw);
      s += qv[qv_index(wi, l, 30 + h * 5 + d)] * wsl[rh * 2 + rw];
    }
  vp[idx] = s;
}

// ---------------------------------------------------------------------------
// Windowed attention with WMMA.  One block per (window, head).
//   corr(256x64) = q(256x5) * vp^T(5x64) / 5 + rpb   then  x_sp = corr * vp
// x_sp goes into columns 0..29 of the stride-60 ybuf.
// ---------------------------------------------------------------------------
__global__ __launch_bounds__(256) void k_attn(const float* __restrict__ qv,
                                              const float* __restrict__ vp,
                                              const float* __restrict__ rpb,
                                              float* __restrict__ ybuf) {
  const int wi = blockIdx.x;
  const int hq = blockIdx.y;
  const int tid = threadIdx.x;
  const int lane = tid & 31;
  const int wave = tid >> 5;
  const int halfHi = (lane >= 16) ? 1 : 0;
  const int mlane = lane & 15;
  __shared__ float vp_s8[MLEN * 8];    // vp rows, K padded 5->8 with zeros
  __shared__ float vpT_s[MLEN * 16];   // vp transposed, N padded 5->16 w/ zeros
  __shared__ float corr_s[8][16 * 64]; // wave-private corr tiles

  const int vbase = (wi * NH + hq) * (MLEN * HD);
  for (int t = tid; t < MLEN * 8; t += 256) {
    const int col = t >> 3, k = t & 7;
    const int kk = (k < 5) ? k : 0;
    vp_s8[t] = (k < 5) ? vp[vbase + col * 5 + kk] : 0.f;
  }
  for (int t = tid; t < MLEN * 16; t += 256) {
    const int k = t >> 4, n = t & 15;
    const int nn = (n < 5) ? n : 0;
    vpT_s[t] = (n < 5) ? vp[vbase + k * 5 + nn] : 0.f;
  }
  __syncthreads();

  const v8f vzero = {0.f, 0.f, 0.f, 0.f, 0.f, 0.f, 0.f, 0.f};

  for (int mt = wave; mt < 16; mt += 8) {
    const int l = mt * 16 + mlane;
    float q0, q1, q2, q3, q4;
    {
      const int base = qv_index(wi, l, hq * 5);
      q0 = qv[base]; q1 = qv[base + 1]; q2 = qv[base + 2];
      q3 = qv[base + 3]; q4 = qv[base + 4];
    }
    // A fragments for K-step 0 (k 0..3) and step 1 (k 4..7, zero padded)
    v2f aS0 = {halfHi ? q2 : q0, halfHi ? q3 : q1};
    v2f aS1 = {halfHi ? 0.f : q4, 0.f};
    const int koff = halfHi ? 2 : 0;

    v8f acc[4];
#pragma unroll
    for (int nt = 0; nt < 4; nt++) {
      acc[nt] = vzero;
      const int col = nt * 16 + mlane;  // vp row index for this B column
      v2f b0 = {vp_s8[col * 8 + koff], vp_s8[col * 8 + koff + 1]};
      acc[nt] = wmma4(aS0, b0, acc[nt]);
      v2f b1 = {vp_s8[col * 8 + 4 + koff], vp_s8[col * 8 + 4 + koff + 1]};
      acc[nt] = wmma4(aS1, b1, acc[nt]);
    }

    // write corr tile (scaled + bias) to this wave's LDS region
    float* cs = corr_s[wave];
#pragma unroll
    for (int nt = 0; nt < 4; nt++) {
#pragma unroll
      for (int i = 0; i < 8; i++) {
        const int mrow = i + (halfHi ? 8 : 0);
        const int mcol = nt * 16 + mlane;
        cs[mrow * 64 + mcol] =
            acc[nt][i] * 0.2f + rpb[(hq * 256 + mt * 16 + mrow) * 64 + mcol];
      }
    }
    // same-wave DS ops are in order: safe to read back without a barrier
    v8f acc2 = vzero;
    for (int ks = 0; ks < 16; ks++) {
      const int k0 = ks * 4 + koff;
      v2f a = {cs[mlane * 64 + k0], cs[mlane * 64 + k0 + 1]};
      v2f b = {vpT_s[k0 * 16 + mlane], vpT_s[(k0 + 1) * 16 + mlane]};
      acc2 = wmma4(a, b, acc2);
    }
#pragma unroll
    for (int i = 0; i < 8; i++) {
      const int mrow = i + (halfHi ? 8 : 0);
      if (mlane < 5)
        ybuf[(wi * LWIN + mt * 16 + mrow) * 60 + hq * 5 + mlane] = acc2[i];
    }
  }
}

// ---------------------------------------------------------------------------
// Channel self-correlation with WMMA
// ---------------------------------------------------------------------------
// corr_c(32x32 padded) = qc^T(30x256) * vc(256x30) / 256 — one block per window
__global__ __launch_bounds__(128) void k_chan(const float* __restrict__ qv,
                                              float* __restrict__ cc) {
  const int wi = blockIdx.x;
  const int tid = threadIdx.x;
  __shared__ float qv_s[LWIN * 62];  // stride 62, cols 60..61 zero (63488 B)
  for (int t = tid; t < LWIN * 62; t += 128) {
    const int l = t / 62, c = t % 62;
    const int ccl = (c < 60) ? c : 0;
    qv_s[t] = (c < 60) ? qv[qv_index(wi, l, ccl)] : 0.f;
  }
  __syncthreads();
  const int lane = tid & 31, wave = tid >> 5;
  const int mt = wave >> 1, nt = wave & 1;
  const int halfHi = (lane >= 16) ? 1 : 0;
  const int mlane = lane & 15;
  const int m = mt * 16 + mlane;   // qc channel (pad >=30 -> zero via select)
  const int n = nt * 16 + mlane;   // vc channel (30+n <= 61: zero padded)
  const bool mv = (m < 30);
  const v8f vzero = {0.f, 0.f, 0.f, 0.f, 0.f, 0.f, 0.f, 0.f};
  v8f acc = vzero;
  for (int ks = 0; ks < 64; ks++) {
    const int k0 = ks * 4 + (halfHi ? 2 : 0);
    v2f a = {mv ? qv_s[k0 * 62 + m] : 0.f,
             mv ? qv_s[(k0 + 1) * 62 + m] : 0.f};
    v2f b = {qv_s[k0 * 62 + 30 + n], qv_s[(k0 + 1) * 62 + 30 + n]};
    acc = wmma4(a, b, acc);
  }
#pragma unroll
  for (int i = 0; i < 8; i++) {
    const int mr = mt * 16 + i + (halfHi ? 8 : 0);
    cc[wi * 1024 + mr * 32 + n] = acc[i] * (1.f / 256.f);
  }
}

// x_ch(256x30) = vc(256x30) * corr_c^T -> columns 30..59 of ybuf
__global__ __launch_bounds__(256) void k_xch(const float* __restrict__ qv,
                                             const float* __restrict__ cc,
                                             float* __restrict__ ybuf) {
  const int wi = blockIdx.x;
  const int tid = threadIdx.x;
  __shared__ float cc_s[1024];
  __shared__ float vc_s[LWIN * 32];  // stride 32, cols 30..31 zero
  for (int t = tid; t < 1024; t += 256) cc_s[t] = cc[wi * 1024 + t];
  for (int t = tid; t < LWIN * 32; t += 256) {
    const int l = t >> 5, c = t & 31;
    const int ccl = (c < 30) ? c : 0;
    vc_s[t] = (c < 30) ? qv[qv_index(wi, l, 30 + ccl)] : 0.f;
  }
  __syncthreads();
  const int lane = tid & 31, wave = tid >> 5;
  const int halfHi = (lane >= 16) ? 1 : 0;
  const int mlane = lane & 15;
  const v8f vzero = {0.f, 0.f, 0.f, 0.f, 0.f, 0.f, 0.f, 0.f};
  for (int mt = wave; mt < 16; mt += 8) {
    v8f acc0 = vzero, acc1 = vzero;
    for (int ks = 0; ks < 8; ks++) {
      const int k0 = ks * 4 + (halfHi ? 2 : 0);
      v2f a = {vc_s[(mt * 16 + mlane) * 32 + k0],
               vc_s[(mt * 16 + mlane) * 32 + k0 + 1]};
      v2f b0 = {cc_s[mlane * 32 + k0], cc_s[mlane * 32 + k0 + 1]};
      acc0 = wmma4(a, b0, acc0);
      v2f b1 = {cc_s[(16 + mlane) * 32 + k0], cc_s[(16 + mlane) * 32 + k0 + 1]};
      acc1 = wmma4(a, b1, acc1);
    }
#pragma unroll
    for (int i = 0; i < 8; i++) {
      const int l = mt * 16 + i + (halfHi ? 8 : 0);
      ybuf[(wi * LWIN + l) * 60 + 30 + mlane] = acc0[i];
      if (mlane < 14) ybuf[(wi * LWIN + l) * 60 + 46 + mlane] = acc1[i];
    }
  }
}

// ---------------------------------------------------------------------------
// Window-reverse + 60x60 projection with WMMA
// ---------------------------------------------------------------------------
__global__ __launch_bounds__(256) void k_proj(const float* __restrict__ ybuf,
                                              const float* __restrict__ wproj,
                                              const float* __restrict__ bproj,
                                              float* __restrict__ out) {
  __shared__ float w_s[64 * 60];  // rows 60..63 zero
  __shared__ float b_s[64];
  const int tid = threadIdx.x;
  for (int t = tid; t < 64 * 60; t += 256) {
    const int n = t / 60, k = t % 60;
    const int nn = (n < 60) ? n : 0;
    w_s[t] = (n < 60) ? wproj[nn * 60 + k] : 0.f;
  }
  if (tid < 64) b_s[tid] = (tid < 60) ? bproj[tid] : 0.f;
  __syncthreads();
  const int lane = tid & 31, wave = tid >> 5;
  const int halfHi = (lane >= 16) ? 1 : 0;
  const int mlane = lane & 15;
  const int tile = blockIdx.x * 8 + wave;  // 9216 M-tiles of 16 pixels
  const int p = tile * 16 + mlane;
  const int b = p / (IMG * IMG);
  const int rem = p % (IMG * IMG);
  const int hh = rem / IMG, ww = rem % IMG;
  const int wi = (b * NWH + hh / HS) * NWH + ww / WS;
  const int l = (hh & 15) * 16 + (ww & 15);
  const int rowbase = (wi * LWIN + l) * 60;
  const v8f vzero = {0.f, 0.f, 0.f, 0.f, 0.f, 0.f, 0.f, 0.f};
  v8f acc[4];
#pragma unroll
  for (int nt = 0; nt < 4; nt++) acc[nt] = vzero;
  for (int ks = 0; ks < 15; ks++) {
    const int k0 = ks * 4 + (halfHi ? 2 : 0);
    v2f a = {ybuf[rowbase + k0], ybuf[rowbase + k0 + 1]};
#pragma unroll
    for (int nt = 0; nt < 4; nt++) {
      const int n = nt * 16 + mlane;
      v2f bb = {w_s[n * 60 + k0], w_s[n * 60 + k0 + 1]};
      acc[nt] = wmma4(a, bb, acc[nt]);
    }
  }
#pragma unroll
  for (int nt = 0; nt < 4; nt++) {
    const int n = nt * 16 + mlane;
    if (n < 60) {
#pragma unroll
      for (int i = 0; i < 8; i++) {
        const int m = i + (halfHi ? 8 : 0);
        out[(tile * 16 + m) * 60 + n] = acc[nt][i] + b_s[n];
      }
    }
  }
}

// ---------------------------------------------------------------------------
// Launch
// ---------------------------------------------------------------------------
extern "C" void kernel_launch(void* const* d_in, const int* in_sizes, int n_in,
                              void* d_out, int out_size, void* d_ws,
                              size_t ws_size, hipStream_t stream) {
  (void)in_sizes; (void)n_in; (void)out_size; (void)ws_size;
  const float* x      = (const float*)d_in[0];
  const float* ref    = (const float*)d_in[1];
  const float* w_c1   = (const float*)d_in[2];
  const float* b_c1   = (const float*)d_in[3];
  const float* w_c2   = (const float*)d_in[4];
  const float* b_c2   = (const float*)d_in[5];
  const float* w_c3   = (const float*)d_in[6];
  const float* b_c3   = (const float*)d_in[7];
  const float* w_lin  = (const float*)d_in[8];
  const float* b_lin  = (const float*)d_in[9];
  const float* w_ra1  = (const float*)d_in[10];
  const float* b_ra1  = (const float*)d_in[11];
  const float* w_ra2  = (const float*)d_in[12];
  const float* b_ra2  = (const float*)d_in[13];
  const float* w_gate = (const float*)d_in[14];
  const float* b_gate = (const float*)d_in[15];
  const float* w_pp   = (const float*)d_in[16];
  const float* b_pp   = (const float*)d_in[17];
  const float* ln1_g  = (const float*)d_in[18];
  const float* ln1_b  = (const float*)d_in[19];
  const float* w_p1   = (const float*)d_in[20];
  const float* b_p1   = (const float*)d_in[21];
  const float* ln2_g  = (const float*)d_in[22];
  const float* ln2_b  = (const float*)d_in[23];
  const float* w_p2   = (const float*)d_in[24];
  const float* b_p2   = (const float*)d_in[25];
  const float* ln3_g  = (const float*)d_in[26];
  const float* ln3_b  = (const float*)d_in[27];
  const float* w_p3   = (const float*)d_in[28];
  const float* b_p3   = (const float*)d_in[29];
  const float* w_sl   = (const float*)d_in[30];
  const float* b_sl   = (const float*)d_in[31];
  const float* w_proj = (const float*)d_in[32];
  const float* b_proj = (const float*)d_in[33];
  float* out = (float*)d_out;

  // workspace carve-out (floats), lifetime-based aliasing
  float* ws   = (float*)d_ws;
  float* c1b  = ws;                    //  1,769,472  (NPIX*12)
  float* c2b  = ws + 1769472;          //  1,769,472  (NPIX*12)
  float* catb = ws + 3538944;          // 17,694,720  (NPIX*120: dfe | ar)
  float* ar1b = ws + 21233664;         //  8,847,360  (NPIX*60)
  float* ptab = ws + 30081024;         //      6,144
  float* rpbb = ws + 30087168;         //     98,304
  float* qvb  = ar1b;                  // alias: ar1 dead after conv ra2
  float* vpb  = c2b;                   // alias: c2 dead after k_dfe
  float* ybuf = catb;                  // alias: cat dead after k_gate
  float* ccb  = c1b;                   // alias: c1 dead after k_c2

  k_c1<<<NPIX / 256, 256, 0, stream>>>(x, w_c1, b_c1, c1b);
  k_c2<<<NPIX / 256, 256, 0, stream>>>(c1b, w_c2, b_c2, c2b);
  k_dfe<<<NPIX * CH / 256, 256, 0, stream>>>(x, c2b, w_c3, b_c3, w_lin, b_lin,
                                             catb);
  // ra1: ref (NCHW) -> ar1 (NHWC) with lrelu;  ra2: ar1 -> cat cols 60..119
  k_conv3x3<true, true><<<NPIX / 128, 256, 0, stream>>>(ref, w_ra1, b_ra1, ar1b,
                                                        60, 0);
  k_conv3x3<false, false><<<NPIX / 128, 256, 0, stream>>>(ar1b, w_ra2, b_ra2,
                                                          catb, 120, 60);
  k_gate<<<NPIX / 128, 256, 0, stream>>>(catb, w_gate, b_gate, qvb);

  k_ptab<<<1, 1024, 0, stream>>>(w_pp, b_pp, ln1_g, ln1_b, w_p1, b_p1, ln2_g,
                                 ln2_b, w_p2, b_p2, ln3_g, ln3_b, w_p3, b_p3,
                                 ptab);
  k_rpb<<<NH * LWIN * MLEN / 256, 256, 0, stream>>>(ptab, rpbb);
  k_vp<<<NWIN * NH * MLEN * HD / 256, 256, 0, stream>>>(qvb, w_sl, b_sl, vpb);

  dim3 gAttn(NWIN, NH);
  k_attn<<<gAttn, 256, 0, stream>>>(qvb, vpb, rpbb, ybuf);
  k_chan<<<NWIN, 128, 0, stream>>>(qvb, ccb);
  k_xch<<<NWIN, 256, 0, stream>>>(qvb, ccb, ybuf);
  k_proj<<<NPIX / 128, 256, 0, stream>>>(ybuf, w_proj, b_proj, out);
}